// Adaptive_Style_Memory_Bank_27848567947833
// MI455X (gfx1250) — compile-verified
//
#include <hip/hip_runtime.h>

typedef float v2f __attribute__((ext_vector_type(2)));
typedef float v8f __attribute__((ext_vector_type(8)));

#define D_DIM 524288
#define M_DIM 256
#define B_DIM 64
#define S_DIM 2048
#define KSLABS 64
#define KSLAB (D_DIM / KSLABS) /* 8192 */

__global__ void zero_ws_kernel(float* __restrict__ ws, int n) {
    int i = blockIdx.x * blockDim.x + threadIdx.x;
    if (i < n) ws[i] = 0.0f;
}

// Split-K WMMA GEMM: dot[b,m] += A[b, kslab] * B[kslab, m]; norm2[m] += sum B^2.
// grid = (16 m-tiles, 64 k-slabs); block = 128 threads = 4 waves (one 16-row
// b-tile per wave, covering all 64 query rows).
__global__ __launch_bounds__(128)
void gemm_splitk_kernel(const float* __restrict__ A,    // [64, 524288]
                        const float* __restrict__ Bm,   // [524288, 256] row-major
                        float* __restrict__ dot,        // [64, 256] accum
                        float* __restrict__ norm2) {    // [256] accum
    const int mtile = blockIdx.x;          // 0..15
    const int slab  = blockIdx.y;          // 0..63
    const int wave  = threadIdx.x >> 5;    // 0..3  -> b-row tile
    const int lane  = threadIdx.x & 31;
    const int lrow  = lane & 15;           // row (A) / col (B) within tile
    const int sel   = lane >> 4;           // 0: K={0,1}, 1: K={2,3}
    const int kp    = sel * 2;
    const int m0    = mtile * 16;
    const int b0    = wave * 16;

    const int kbeg = slab * KSLAB;

    // A: lane streams its own row; 8B-aligned pair loads (K pair).
    const float* aptr = A + (size_t)(b0 + lrow) * D_DIM + kbeg + kp;
    // B: lanes 0..15 read 64 contiguous bytes of row (kbeg+kp); +M_DIM for K+1.
    const float* bptr = Bm + (size_t)(kbeg + kp) * M_DIM + m0 + lrow;

    v8f c0 = {};
    v8f c1 = {};
    float nacc = 0.0f;

    // KSLAB/8 iterations; two independent WMMA accumulator chains per iter.
    for (int k = 0; k < KSLAB; k += 8) {
        v2f a0 = *(const v2f*)(aptr);
        v2f a1 = *(const v2f*)(aptr + 4);
        float bv0 = bptr[0];
        float bv1 = bptr[M_DIM];
        float bv2 = bptr[4 * M_DIM];
        float bv3 = bptr[5 * M_DIM];
        v2f b0v = { bv0, bv1 };
        v2f b1v = { bv2, bv3 };

        c0 = __builtin_amdgcn_wmma_f32_16x16x4_f32(false, a0, false, b0v,
                                                   (short)0, c0, false, false);
        c1 = __builtin_amdgcn_wmma_f32_16x16x4_f32(false, a1, false, b1v,
                                                   (short)0, c1, false, false);

        nacc += bv0 * bv0 + bv1 * bv1 + bv2 * bv2 + bv3 * bv3;

        aptr += 8;
        bptr += (size_t)8 * M_DIM;
    }

    // C layout (32-bit 16x16): VGPR r holds row r (lanes 0-15) / row r+8
    // (lanes 16-31); column = lane & 15.
    const int col = m0 + lrow;
    #pragma unroll
    for (int r = 0; r < 8; ++r) {
        int row = b0 + r + sel * 8;
        float v = (c0[r] + c1[r]);
        atomicAdd(&dot[row * M_DIM + col], v);
    }
    // norm2: only one wave contributes (B tile identical across waves).
    // Lanes L and L+16 both hold column (m0 + lrow) partials.
    if (wave == 0) {
        atomicAdd(&norm2[col], nacc);
    }
}

// One block per query row: sim = dot/max(sqrt(norm2),eps), first-max argmax,
// then gather bank_style[idx] into out.
__global__ __launch_bounds__(256)
void argmax_gather_kernel(const float* __restrict__ dot,
                          const float* __restrict__ norm2,
                          const float* __restrict__ style,  // [256, 2048]
                          float* __restrict__ out) {        // [64, 2048]
    __shared__ float sval[256];
    __shared__ int   sidx[256];
    const int b = blockIdx.x;
    const int m = threadIdx.x;

    float nrm = sqrtf(norm2[m]);
    nrm = fmaxf(nrm, 1e-12f);
    sval[m] = dot[b * M_DIM + m] / nrm;
    sidx[m] = m;
    __syncthreads();

    // Tree argmax with lower-index-wins tie-break (matches jnp.argmax).
    for (int s = 128; s > 0; s >>= 1) {
        if (m < s) {
            float vo = sval[m + s];
            int   io = sidx[m + s];
            if (vo > sval[m] || (vo == sval[m] && io < sidx[m])) {
                sval[m] = vo;
                sidx[m] = io;
            }
        }
        __syncthreads();
    }
    const int idx = sidx[0];

    for (int i = threadIdx.x; i < S_DIM; i += 256) {
        out[(size_t)b * S_DIM + i] = style[(size_t)idx * S_DIM + i];
    }
}

extern "C" void kernel_launch(void* const* d_in, const int* in_sizes, int n_in,
                              void* d_out, int out_size, void* d_ws, size_t ws_size,
                              hipStream_t stream) {
    const float* content      = (const float*)d_in[0]; // [64,512,32,32]
    const float* bank_content = (const float*)d_in[1]; // [524288,256]
    const float* bank_style   = (const float*)d_in[2]; // [256,2048]
    float* out = (float*)d_out;

    float* dot   = (float*)d_ws;            // 64*256 floats
    float* norm2 = dot + B_DIM * M_DIM;     // 256 floats

    const int nzero = B_DIM * M_DIM + M_DIM;
    zero_ws_kernel<<<(nzero + 255) / 256, 256, 0, stream>>>(dot, nzero);

    gemm_splitk_kernel<<<dim3(16, KSLABS), 128, 0, stream>>>(
        content, bank_content, dot, norm2);

    argmax_gather_kernel<<<B_DIM, 256, 0, stream>>>(dot, norm2, bank_style, out);
}